// MaximumLikelihoodDetector_63952063037692
// MI455X (gfx1250) — compile-verified
//
#include <hip/hip_runtime.h>
#include <hip/hip_bf16.h>
#include <math.h>

typedef __attribute__((ext_vector_type(2))) float v2f;
typedef __attribute__((ext_vector_type(8))) float v8f;

#define MM 8      // antennas
#define KK 3      // streams
#define PP 16     // constellation size
#define VV 4096   // candidates = PP^KK
#define NBB 4     // bits per symbol

__global__ __launch_bounds__(256)
void mld_wmma_kernel(const float* __restrict__ y_real, const float* __restrict__ y_imag,
                     const float* __restrict__ h_real, const float* __restrict__ h_imag,
                     const float* __restrict__ s_real, const float* __restrict__ s_imag,
                     const float* __restrict__ vecs_real, const float* __restrict__ vecs_imag,
                     const int* __restrict__ c1, const int* __restrict__ c0,
                     float* __restrict__ out) {
    const int b   = blockIdx.x;
    const int tid = threadIdx.x;

    // Augmented real operator with y folded in:
    //   Ha(16x8): rows 0..7 = real eqns, rows 8..15 = imag eqns
    //   cols 0..2 pair with Re(x), cols 3..5 pair with Im(x),
    //   col 6 = -y_t (pairs with all-ones X row), col 7 = 0.
    // => D = Ha @ Xaug gives (hx - y_t) directly; metric = sum of squares of column.
    __shared__ float sHa[16][8];
    __shared__ float sLr[8][8], sLi[8][8];
    __shared__ float sGr[8][3], sGi[8][3];
    __shared__ float sZr[8], sZi[8];
    __shared__ float sExps[VV];           // -distance^2 per candidate
    __shared__ float sPart[48][4];
    __shared__ float sGmax[48];
    __shared__ float sLogits[48];         // [kk*16 + s]

    // ---------------- Phase 1: Cholesky whitening (thread 0, serial, tiny) ------------
    if (tid == 0) {
        const float* Sr = s_real + (size_t)b * 64;
        const float* Si = s_imag + (size_t)b * 64;
        for (int j = 0; j < 8; ++j) {
            float sum = Sr[j * 8 + j];
            for (int k = 0; k < j; ++k)
                sum -= sLr[j][k] * sLr[j][k] + sLi[j][k] * sLi[j][k];
            float d = sqrtf(fmaxf(sum, 1e-30f));
            sLr[j][j] = d; sLi[j][j] = 0.0f;
            float inv = 1.0f / d;
            for (int i = j + 1; i < 8; ++i) {
                float ar = Sr[i * 8 + j], ai = Si[i * 8 + j];
                for (int k = 0; k < j; ++k) {
                    float lr = sLr[i][k], li = sLi[i][k];
                    float mr = sLr[j][k], mi = sLi[j][k];
                    ar -= lr * mr + li * mi;     // L[i][k] * conj(L[j][k])
                    ai -= li * mr - lr * mi;
                }
                sLr[i][j] = ar * inv; sLi[i][j] = ai * inv;
            }
        }
        // z = L^{-1} y
        const float* yr = y_real + (size_t)b * 8;
        const float* yi = y_imag + (size_t)b * 8;
        for (int i = 0; i < 8; ++i) {
            float cr = yr[i], ci = yi[i];
            for (int j = 0; j < i; ++j) {
                cr -= sLr[i][j] * sZr[j] - sLi[i][j] * sZi[j];
                ci -= sLr[i][j] * sZi[j] + sLi[i][j] * sZr[j];
            }
            float inv = 1.0f / sLr[i][i];
            sZr[i] = cr * inv; sZi[i] = ci * inv;
        }
        // G = L^{-1} H
        const float* hr = h_real + (size_t)b * 24;
        const float* hi = h_imag + (size_t)b * 24;
        for (int k = 0; k < 3; ++k) {
            for (int i = 0; i < 8; ++i) {
                float cr = hr[i * 3 + k], ci = hi[i * 3 + k];
                for (int j = 0; j < i; ++j) {
                    cr -= sLr[i][j] * sGr[j][k] - sLi[i][j] * sGi[j][k];
                    ci -= sLr[i][j] * sGi[j][k] + sLi[i][j] * sGr[j][k];
                }
                float inv = 1.0f / sLr[i][i];
                sGr[i][k] = cr * inv; sGi[i][k] = ci * inv;
            }
        }
        // [Re(Gx)-Re(z)]   [Gr -Gi -zr][Xr]
        // [Im(Gx)-Im(z)] = [Gi  Gr -zi][Xi]
        //                              [ 1]
        for (int m = 0; m < 8; ++m) {
            for (int k = 0; k < 3; ++k) {
                sHa[m][k]         = sGr[m][k];
                sHa[m][3 + k]     = -sGi[m][k];
                sHa[m + 8][k]     = sGi[m][k];
                sHa[m + 8][3 + k] = sGr[m][k];
            }
            sHa[m][6]     = -sZr[m];  sHa[m][7]     = 0.0f;
            sHa[m + 8][6] = -sZi[m];  sHa[m + 8][7] = 0.0f;
        }
    }
    __syncthreads();

    // ---------------- Phase 2: WMMA sweep over 4096 candidates -------------------------
    // Per 16-candidate tile: D(16x16) = Ha(16x8) @ Xaug(8x16), two f32 16x16x4 WMMAs.
    const int wave = tid >> 5;
    const int lane = tid & 31;
    const int m16  = lane & 15;
    const int half = lane >> 4;

    // A-operand (16x4 f32): lane M = lane%16, VGPR g holds K = 2*(lane/16) + g
    v2f a0, a1;
    a0.x = sHa[m16][2 * half];     a0.y = sHa[m16][2 * half + 1];      // K-chunk 0: cols 0..3
    a1.x = sHa[m16][4 + 2 * half]; a1.y = sHa[m16][4 + 2 * half + 1];  // K-chunk 1: cols 4..7

    // B-operand (4x16 f32): VGPR g holds K = 2*(lane/16) + g, N = lane%16.
    // Xaug rows: 0..2 = Re(vec), 3..5 = Im(vec), 6 = 1.0, 7 = 0.0
    //   half==0 lanes need rows {0,1} then {4,5}: Re0, Re1 | Im1, Im2
    //   half==1 lanes need rows {2,3} then {6,7}: Re2, Im0 | 1.0, 0.0
    const float* pb0x = vecs_real + (half ? 2 : 0);
    const float* pb0y = half ? vecs_imag : (vecs_real + 1);

    for (int t = wave; t < VV / 16; t += 8) {
        const int v  = t * 16 + m16;   // candidate column this lane owns
        const int v3 = v * 3;
        v2f b0, b1;
        b0.x = pb0x[v3];
        b0.y = pb0y[v3];
        float t1 = vecs_imag[v3 + 1];
        float t2 = vecs_imag[v3 + 2];
        b1.x = half ? 1.0f : t1;       // branchless: v_cndmask
        b1.y = half ? 0.0f : t2;

        v8f acc = {};
        acc = __builtin_amdgcn_wmma_f32_16x16x4_f32(false, a0, false, b0,
                                                    (short)0, acc, false, false);
        acc = __builtin_amdgcn_wmma_f32_16x16x4_f32(false, a1, false, b1,
                                                    (short)0, acc, false, false);

        // D rows M = g + 8*half hold (hx - y_t); accumulate squared residual.
        float ssum = acc[0] * acc[0];
        #pragma unroll
        for (int g = 1; g < 8; ++g)
            ssum = fmaf(acc[g], acc[g], ssum);
        ssum += __shfl_xor(ssum, 16);   // combine real+imag halves (wave32)
        if (half == 0) sExps[v] = -ssum;
    }
    __syncthreads();

    // ---------------- Phase 3: logsumexp over 48 groups of 256 -------------------------
    // group g = kk*16 + s ; member i's candidate: v = (i>>4)*W1 + (i&15)*W2 + s*W
    int g = tid >> 2, sub = tid & 3;
    int W = 0, W1 = 0, W2 = 0;
    if (tid < 192) {
        int kk = g >> 4;
        if (kk == 0)      { W = 256; W1 = 16;  W2 = 1;  }
        else if (kk == 1) { W = 16;  W1 = 256; W2 = 1;  }
        else              { W = 1;   W1 = 256; W2 = 16; }
        int s = g & 15;
        float lmax = -3.4e38f;
        for (int u = 0; u < 64; ++u) {
            int i = sub * 64 + u;
            int v = (i >> 4) * W1 + (i & 15) * W2 + s * W;
            lmax = fmaxf(lmax, sExps[v]);
        }
        sPart[g][sub] = lmax;
    }
    __syncthreads();
    if (tid < 48) {
        sGmax[tid] = fmaxf(fmaxf(sPart[tid][0], sPart[tid][1]),
                           fmaxf(sPart[tid][2], sPart[tid][3]));
    }
    __syncthreads();
    if (tid < 192) {
        int s = g & 15;
        float gm = sGmax[g];
        float lsum = 0.0f;
        for (int u = 0; u < 64; ++u) {
            int i = sub * 64 + u;
            int v = (i >> 4) * W1 + (i & 15) * W2 + s * W;
            lsum += expf(sExps[v] - gm);
        }
        sPart[g][sub] = lsum;
    }
    __syncthreads();
    if (tid < 48) {
        sLogits[tid] = sGmax[tid] +
            logf(sPart[tid][0] + sPart[tid][1] + sPart[tid][2] + sPart[tid][3]);
    }
    __syncthreads();

    // ---------------- Phase 4: LLRs = LSE(bit=1 symbols) - LSE(bit=0 symbols) ----------
    if (tid < KK * NBB) {
        int kk = tid >> 2, j = tid & 3;
        float m1 = -3.4e38f, m0 = -3.4e38f;
        #pragma unroll
        for (int u = 0; u < 8; ++u) {
            m1 = fmaxf(m1, sLogits[kk * 16 + c1[j * 8 + u]]);
            m0 = fmaxf(m0, sLogits[kk * 16 + c0[j * 8 + u]]);
        }
        float s1 = 0.0f, s0 = 0.0f;
        #pragma unroll
        for (int u = 0; u < 8; ++u) {
            s1 += expf(sLogits[kk * 16 + c1[j * 8 + u]] - m1);
            s0 += expf(sLogits[kk * 16 + c0[j * 8 + u]] - m0);
        }
        out[b * (KK * NBB) + tid] = (m1 + logf(s1)) - (m0 + logf(s0));
    }
}

extern "C" void kernel_launch(void* const* d_in, const int* in_sizes, int n_in,
                              void* d_out, int out_size, void* d_ws, size_t ws_size,
                              hipStream_t stream) {
    const float* y_real    = (const float*)d_in[0];
    const float* y_imag    = (const float*)d_in[1];
    const float* h_real    = (const float*)d_in[2];
    const float* h_imag    = (const float*)d_in[3];
    const float* s_real    = (const float*)d_in[4];
    const float* s_imag    = (const float*)d_in[5];
    const float* vecs_real = (const float*)d_in[6];
    const float* vecs_imag = (const float*)d_in[7];
    // d_in[8] = c (unused: membership computed from base-16 digits of v)
    const int* c1 = (const int*)d_in[9];
    const int* c0 = (const int*)d_in[10];
    float* out = (float*)d_out;

    const int B = in_sizes[0] / MM;   // 1024
    mld_wmma_kernel<<<dim3(B), dim3(256), 0, stream>>>(
        y_real, y_imag, h_real, h_imag, s_real, s_imag,
        vecs_real, vecs_imag, c1, c0, out);
}